// buff_Loss_29695403885106
// MI455X (gfx1250) — compile-verified
//
#include <hip/hip_runtime.h>

// CDNA5 / gfx1250. wave32. Streaming reduction:
//   mean( (scale(l) * (n - l))^2 ),  scale = {1,5,3}[l]
// Memory-bound: 268MB @ 23.3 TB/s ~= 11.5us floor. WMMA (f32 16x16x4) is used
// for the exact cross-lane wave reduction (A = acc, B = ones -> row sums).

typedef __attribute__((ext_vector_type(2))) float v2f;
typedef __attribute__((ext_vector_type(4))) float v4f;
typedef __attribute__((ext_vector_type(4))) int   v4i;
typedef __attribute__((ext_vector_type(8))) float v8f;

// Exact 32-lane sum via V_WMMA_F32_16X16X4_F32.
// A (16x4 f32, 2 VGPR): lanes 0-15 vgpr0 = A[m][0], lanes 16-31 vgpr0 = A[m][2].
// Put acc in vgpr0, 0 in vgpr1; B = all ones. Then D[m][j] = acc[m]+acc[m+16].
// Per-lane sum of the 8 D VGPRs gives half-sum; xor-16 shuffle completes it.
// EXEC must be all ones at the call site (no divergence here).
__device__ __forceinline__ float wave_sum_wmma(float acc) {
  v2f a; a[0] = acc;  a[1] = 0.0f;
  v2f b; b[0] = 1.0f; b[1] = 1.0f;
  v8f c = {};
  v8f d = __builtin_amdgcn_wmma_f32_16x16x4_f32(
      /*neg_a=*/false, a, /*neg_b=*/false, b,
      /*c_mod=*/(short)0, c, /*reuse_a=*/false, /*reuse_b=*/false);
  float h = ((d[0] + d[1]) + (d[2] + d[3])) + ((d[4] + d[5]) + (d[6] + d[7]));
  h += __shfl_xor(h, 16, 32);   // combine rows 0-7 half with rows 8-15 half
  return h;                     // every lane holds the full 32-lane sum
}

__device__ __forceinline__ float sq_term(float n, int lab) {
  // branchless: scale = 1 | 5 | 3
  float s = (lab == 1) ? 5.0f : ((lab == 2) ? 3.0f : 1.0f);
  float diff = s * (n - (float)lab);
  return diff * diff;
}

__global__ void __launch_bounds__(256)
loss_partial_kernel(const float* __restrict__ norms,
                    const int*   __restrict__ labels,
                    float*       __restrict__ partials,
                    long long n) {
  const long long nvec   = n >> 2;                       // float4 count
  const long long tid    = (long long)blockIdx.x * blockDim.x + threadIdx.x;
  const long long stride = (long long)gridDim.x * blockDim.x;

  const v4f* __restrict__ nv = (const v4f*)norms;
  const v4i* __restrict__ lv = (const v4i*)labels;

  float acc = 0.0f;

  // 2x unrolled grid-stride loop over 128-bit packets; NT hints (read-once
  // stream, 256MB > 192MB L2 -> don't cache).
  long long i = tid;
  for (; i + stride < nvec; i += 2 * stride) {
    v4f x0 = __builtin_nontemporal_load(nv + i);
    v4i l0 = __builtin_nontemporal_load(lv + i);
    v4f x1 = __builtin_nontemporal_load(nv + i + stride);
    v4i l1 = __builtin_nontemporal_load(lv + i + stride);
#pragma unroll
    for (int k = 0; k < 4; ++k) acc += sq_term(x0[k], l0[k]);
#pragma unroll
    for (int k = 0; k < 4; ++k) acc += sq_term(x1[k], l1[k]);
  }
  for (; i < nvec; i += stride) {
    v4f x = __builtin_nontemporal_load(nv + i);
    v4i l = __builtin_nontemporal_load(lv + i);
#pragma unroll
    for (int k = 0; k < 4; ++k) acc += sq_term(x[k], l[k]);
  }
  // scalar tail (n not multiple of 4; not hit for 4096*8192)
  for (long long j = (nvec << 2) + tid; j < n; j += stride)
    acc += sq_term(norms[j], labels[j]);

  // converged here: EXEC all ones -> WMMA wave reduction
  float wsum = wave_sum_wmma(acc);

  __shared__ float lds[8];            // 256 threads / wave32 = 8 waves
  const int lane = threadIdx.x & 31;
  const int wid  = threadIdx.x >> 5;
  if (lane == 0) lds[wid] = wsum;
  __syncthreads();
  if (threadIdx.x == 0) {
    float bsum = 0.0f;
#pragma unroll
    for (int w = 0; w < 8; ++w) bsum += lds[w];
    partials[blockIdx.x] = bsum;      // deterministic: no atomics
  }
}

__global__ void __launch_bounds__(256)
loss_final_kernel(const float* __restrict__ partials, int numPartials,
                  float* __restrict__ out, float invN) {
  float acc = 0.0f;
  for (int i = threadIdx.x; i < numPartials; i += 256)
    acc += partials[i];

  float wsum = wave_sum_wmma(acc);

  __shared__ float lds[8];
  const int lane = threadIdx.x & 31;
  const int wid  = threadIdx.x >> 5;
  if (lane == 0) lds[wid] = wsum;
  __syncthreads();
  if (threadIdx.x == 0) {
    float s = 0.0f;
#pragma unroll
    for (int w = 0; w < 8; ++w) s += lds[w];
    out[0] = s * invN;
  }
}

extern "C" void kernel_launch(void* const* d_in, const int* in_sizes, int n_in,
                              void* d_out, int out_size, void* d_ws, size_t ws_size,
                              hipStream_t stream) {
  const float* norms  = (const float*)d_in[0];
  const int*   labels = (const int*)d_in[1];
  float* out      = (float*)d_out;
  float* partials = (float*)d_ws;

  const long long n = (long long)in_sizes[0];   // 4096 * 8192 = 33,554,432

  int blocks = 2048;                             // 524288 threads -> 16 f4/i4 pairs each
  long long maxB = (long long)(ws_size / sizeof(float));
  if (maxB < 1) maxB = 1;
  if ((long long)blocks > maxB) blocks = (int)maxB;

  loss_partial_kernel<<<blocks, 256, 0, stream>>>(norms, labels, partials, n);

  const float invN = (float)(1.0 / (double)n);
  loss_final_kernel<<<1, 256, 0, stream>>>(partials, blocks, out, invN);
}